// PositionalSparseLinear_79121887527366
// MI455X (gfx1250) — compile-verified
//
#include <hip/hip_runtime.h>
#include <stdint.h>
#include <stddef.h>

// Problem constants (match reference)
#define BATCH 1024
#define IN_F  8192
#define OUT_F 8192
#define KPO   32

// Transpose tile: 32 batch-rows x 64 feature-cols of f32 (8 KB payload).
// TDM pads LDS rows: +2 dwords every 64 dwords -> pitch 66 dwords.
#define TR_ROWS  32
#define TR_COLS  64
#define TR_PITCH 66

typedef unsigned int u32x4 __attribute__((ext_vector_type(4)));
typedef int          i32x8 __attribute__((ext_vector_type(8)));
typedef int          i32x4 __attribute__((ext_vector_type(4)));
typedef float        v2f   __attribute__((ext_vector_type(2)));
typedef float        v8f   __attribute__((ext_vector_type(8)));

// ---------------------------------------------------------------------------
// Kernel 1: x[B, IN] (row-major) -> xT[IN, B].
//  - Tensor Data Mover streams the strided 32x64 tile into LDS (TENSORcnt).
//  - Each wave transposes one 16x16 subtile on the matrix pipe:
//      D = sum_kb A_kb * I_kb  with A = subtile (A-layout: M in lanes, K in
//      VGPRs) and D in C-layout (N in lanes, M in VGPRs) -> transposed
//      orientation for free; lanes then store 8 contiguous batches/feature.
// ---------------------------------------------------------------------------
__global__ __launch_bounds__(256) void psl_transpose(const float* __restrict__ x,
                                                     float* __restrict__ xT) {
  __shared__ float tile[TR_ROWS * TR_PITCH];
  const int col_base = blockIdx.x * TR_COLS;   // feature
  const int row_base = blockIdx.y * TR_ROWS;   // batch

#if __has_builtin(__builtin_amdgcn_tensor_load_to_lds)
  if (threadIdx.x < 32) {  // wave 0 issues the TDM op once (EXEC ignored by TDM)
    const uint64_t gaddr = (uint64_t)(const void*)(x + (size_t)row_base * IN_F + col_base);
    const unsigned laddr = (unsigned)(size_t)(void*)tile;  // low 32b of generic = LDS offset

    u32x4 g0;
    g0[0] = 1u;                                                   // count=1, user desc, no gather
    g0[1] = laddr;                                                // lds_addr
    g0[2] = (unsigned)gaddr;                                      // global_addr[31:0]
    g0[3] = ((unsigned)(gaddr >> 32) & 0x01FFFFFFu) | (2u << 30); // global_addr[56:32] | type=2

    i32x8 g1;
    g1[0] = (2 << 16)     // data_size = 4 bytes
          | (1 << 20)     // pad_enable
          | (5 << 22)     // pad_interval: every 64 dwords (256B row)
          | (1 << 25);    // pad_amount: 2 dwords -> LDS pitch = 66 dwords
    g1[1] = (IN_F & 0xFFFF) << 16;                    // tensor_dim0 lo16
    g1[2] = (IN_F >> 16) | ((BATCH & 0xFFFF) << 16);  // tensor_dim0 hi | tensor_dim1 lo
    g1[3] = (BATCH >> 16) | (TR_COLS << 16);          // tensor_dim1 hi | tile_dim0
    g1[4] = TR_ROWS;                                  // tile_dim1 (tile_dim2 = 0)
    g1[5] = IN_F;                                     // tensor_dim0_stride lo32 (elements)
    g1[6] = 0;
    g1[7] = 0;

    i32x4 gz4 = {0, 0, 0, 0};                 // groups 2/3 unused (2D tensor)
    i32x8 gz8 = {0, 0, 0, 0, 0, 0, 0, 0};     // trailing group (clang-23 6-arg form)
    __builtin_amdgcn_tensor_load_to_lds(g0, g1, gz4, gz4, gz8, 0);
    __builtin_amdgcn_s_wait_tensorcnt(0);
  }
  __syncthreads();
#else
  // Fallback: cooperative vector loads into the padded tile.
  {
    const int r  = threadIdx.x >> 3;        // 0..31
    const int cq = (threadIdx.x & 7) * 8;   // 0,8,...,56
    const float* src = x + (size_t)(row_base + r) * IN_F + col_base + cq;
    const float4 a = *(const float4*)(src);
    const float4 b = *(const float4*)(src + 4);
    float* t = &tile[r * TR_PITCH + cq];
    t[0] = a.x; t[1] = a.y; t[2] = a.z; t[3] = a.w;
    t[4] = b.x; t[5] = b.y; t[6] = b.z; t[7] = b.w;
  }
  __syncthreads();
#endif

#if __has_builtin(__builtin_amdgcn_wmma_f32_16x16x4_f32)
  // ---- WMMA layout-rotation transpose: one 16x16 subtile per wave ----
  const int wave = threadIdx.x >> 5;
  const int lane = threadIdx.x & 31;
  const int sc   = (wave & 3) * 16;     // feature offset within tile (0..48)
  const int sr   = (wave >> 2) * 16;    // batch   offset within tile (0 or 16)
  const int nm   = lane & 15;           // A: row M ; B: col N (same lane mapping)
  const int kh   = (lane >> 4) * 2;     // K pair held by this half-wave (0 or 2)

  v8f acc = {0.f, 0.f, 0.f, 0.f, 0.f, 0.f, 0.f, 0.f};
#pragma unroll
  for (int kb = 0; kb < 4; ++kb) {
    // A fragment: S[m, 4*kb + kh + {0,1}] -- row-contiguous ds_load_b64.
    const v2f a = *(const v2f*)&tile[(sr + nm) * TR_PITCH + sc + kb * 4 + kh];
    // B fragment: identity slice I[4*kb + k', n] built in registers.
    v2f b;
    b.x = (kb * 4 + kh + 0 == nm) ? 1.f : 0.f;
    b.y = (kb * 4 + kh + 1 == nm) ? 1.f : 0.f;
    acc = __builtin_amdgcn_wmma_f32_16x16x4_f32(
        false, a, false, b, (short)0, acc, false, false);
  }
  // D in C-layout: lane holds feature n = nm, VGPR i holds batch m = i + (lane>>4)*8.
  float* dst = xT + (size_t)(col_base + sc + nm) * BATCH
                  + row_base + sr + (lane >> 4) * 8;
  float4 lo, hi;
  lo.x = acc[0]; lo.y = acc[1]; lo.z = acc[2]; lo.w = acc[3];
  hi.x = acc[4]; hi.y = acc[5]; hi.z = acc[6]; hi.w = acc[7];
  *(float4*)(dst)     = lo;
  *(float4*)(dst + 4) = hi;
#else
  // Fallback: column reads from the padded tile.
  const int c  = threadIdx.x >> 2;         // 0..63 feature within tile
  const int rq = (threadIdx.x & 3) * 8;    // 0,8,16,24 batch within tile
  float4 a, b;
  a.x = tile[(rq + 0) * TR_PITCH + c];
  a.y = tile[(rq + 1) * TR_PITCH + c];
  a.z = tile[(rq + 2) * TR_PITCH + c];
  a.w = tile[(rq + 3) * TR_PITCH + c];
  b.x = tile[(rq + 4) * TR_PITCH + c];
  b.y = tile[(rq + 5) * TR_PITCH + c];
  b.z = tile[(rq + 6) * TR_PITCH + c];
  b.w = tile[(rq + 7) * TR_PITCH + c];
  float* dst = xT + (size_t)(col_base + c) * BATCH + row_base + rq;
  *(float4*)(dst)     = a;
  *(float4*)(dst + 4) = b;
#endif
}

// ---------------------------------------------------------------------------
// Kernel 2: out[b,o] = sum_k xT[conn[o,k], b] * w[o,k]
// Wave = 1 output x 128 batches; lane k caches conn/w, broadcast via __shfl.
// Gathered loads are coalesced b128 (512B per wave-load) out of L2-resident xT.
// ---------------------------------------------------------------------------
__global__ __launch_bounds__(256) void psl_main(const float* __restrict__ xT,
                                                const int* __restrict__ conn,
                                                const float* __restrict__ wgt,
                                                float* __restrict__ out) {
  __shared__ float so[128][9];  // 128 local batches x 8 outputs (+pad)
  const int lane   = threadIdx.x & 31;
  const int wave   = threadIdx.x >> 5;
  const int o      = blockIdx.x * 8 + wave;
  const int b_base = blockIdx.y * 128;

  const int   ci = conn[o * KPO + lane];  // lane k holds index k (wave32)
  const float wi = wgt [o * KPO + lane];

  float4 acc = make_float4(0.f, 0.f, 0.f, 0.f);
  const float* xb = xT + b_base + lane * 4;
#pragma unroll
  for (int k = 0; k < KPO; ++k) {
    const int   idx = __shfl(ci, k, 32);  // v_readlane broadcast
    const float wk  = __shfl(wi, k, 32);
    const float4 v = *(const float4*)(xb + (size_t)idx * BATCH);
    acc.x = fmaf(wk, v.x, acc.x);
    acc.y = fmaf(wk, v.y, acc.y);
    acc.z = fmaf(wk, v.z, acc.z);
    acc.w = fmaf(wk, v.w, acc.w);
  }

  const int bl = lane * 4;
  so[bl + 0][wave] = acc.x;
  so[bl + 1][wave] = acc.y;
  so[bl + 2][wave] = acc.z;
  so[bl + 3][wave] = acc.w;
  __syncthreads();

  const int row  = threadIdx.x >> 1;        // 0..127
  const int half = (threadIdx.x & 1) * 4;   // 0 or 4
  float4 r;
  r.x = so[row][half + 0];
  r.y = so[row][half + 1];
  r.z = so[row][half + 2];
  r.w = so[row][half + 3];
  *(float4*)(out + (size_t)(b_base + row) * OUT_F + blockIdx.x * 8 + half) = r;
}

// ---------------------------------------------------------------------------
// Fallback (workspace too small): gather straight from row-major x.
// ---------------------------------------------------------------------------
__global__ __launch_bounds__(256) void psl_direct(const float* __restrict__ x,
                                                  const int* __restrict__ conn,
                                                  const float* __restrict__ wgt,
                                                  float* __restrict__ out) {
  __shared__ float so[128][9];
  const int lane   = threadIdx.x & 31;
  const int wave   = threadIdx.x >> 5;
  const int o      = blockIdx.x * 8 + wave;
  const int b_base = blockIdx.y * 128;

  const int   ci = conn[o * KPO + lane];
  const float wi = wgt [o * KPO + lane];

  float4 acc = make_float4(0.f, 0.f, 0.f, 0.f);
  const int b0 = b_base + lane * 4;
#pragma unroll 4
  for (int k = 0; k < KPO; ++k) {
    const int   idx = __shfl(ci, k, 32);
    const float wk  = __shfl(wi, k, 32);
    acc.x = fmaf(wk, x[(size_t)(b0 + 0) * IN_F + idx], acc.x);
    acc.y = fmaf(wk, x[(size_t)(b0 + 1) * IN_F + idx], acc.y);
    acc.z = fmaf(wk, x[(size_t)(b0 + 2) * IN_F + idx], acc.z);
    acc.w = fmaf(wk, x[(size_t)(b0 + 3) * IN_F + idx], acc.w);
  }

  const int bl = lane * 4;
  so[bl + 0][wave] = acc.x;
  so[bl + 1][wave] = acc.y;
  so[bl + 2][wave] = acc.z;
  so[bl + 3][wave] = acc.w;
  __syncthreads();

  const int row  = threadIdx.x >> 1;
  const int half = (threadIdx.x & 1) * 4;
  float4 r;
  r.x = so[row][half + 0];
  r.y = so[row][half + 1];
  r.z = so[row][half + 2];
  r.w = so[row][half + 3];
  *(float4*)(out + (size_t)(b_base + row) * OUT_F + blockIdx.x * 8 + half) = r;
}

extern "C" void kernel_launch(void* const* d_in, const int* in_sizes, int n_in,
                              void* d_out, int out_size, void* d_ws, size_t ws_size,
                              hipStream_t stream) {
  (void)in_sizes; (void)n_in; (void)out_size;
  const float* x    = (const float*)d_in[0];
  const int*   conn = (const int*)d_in[1];
  const float* wgt  = (const float*)d_in[2];
  float*       out  = (float*)d_out;

  const size_t xT_bytes = (size_t)IN_F * BATCH * sizeof(float);  // 32 MB
  if (ws_size >= xT_bytes && d_ws != nullptr) {
    float* xT = (float*)d_ws;
    dim3 tgrid(IN_F / TR_COLS, BATCH / TR_ROWS);  // 128 x 32
    psl_transpose<<<tgrid, 256, 0, stream>>>(x, xT);
    dim3 mgrid(OUT_F / 8, BATCH / 128);           // 1024 x 8
    psl_main<<<mgrid, 256, 0, stream>>>(xT, conn, wgt, out);
  } else {
    dim3 mgrid(OUT_F / 8, BATCH / 128);
    psl_direct<<<mgrid, 256, 0, stream>>>(x, conn, wgt, out);
  }
}